// DAWNLayer_50517405335593
// MI455X (gfx1250) — compile-verified
//
#include <hip/hip_runtime.h>
#include <math.h>

// ---------------------------------------------------------------------------
// Shapes: B=4, S=512, H=512, N=64, P=128, BS = 2048, 4H = 2048, N*H = 32768
// ---------------------------------------------------------------------------

typedef __attribute__((ext_vector_type(2))) float v2f;
typedef __attribute__((ext_vector_type(8))) float v8f;
typedef int v4i_ __attribute__((vector_size(16)));  // matches builtin param type

#if defined(__has_builtin)
#if __has_builtin(__builtin_amdgcn_global_load_async_to_lds_b128) && \
    __has_builtin(__builtin_amdgcn_s_wait_asynccnt)
#define HAVE_ASYNC_LDS 1
#endif
#endif
#ifndef HAVE_ASYNC_LDS
#define HAVE_ASYNC_LDS 0
#endif

__device__ __forceinline__ v8f wmma4(v2f a, v2f b, v8f c) {
  // D = A(16x4) * B(4x16) + C(16x16), fp32
  return __builtin_amdgcn_wmma_f32_16x16x4_f32(false, a, false, b, (short)0, c,
                                               false, false);
}

// A fragment, row-major MxK matrix, 16x4 tile at (m0,k0).
__device__ __forceinline__ v2f load_a_rm(const float* __restrict__ A, int lda,
                                         int m0, int k0, int lane) {
  int row = m0 + (lane & 15);
  int kk  = k0 + ((lane >> 4) << 1);
  v2f a;
  a.x = A[(size_t)row * lda + kk];
  a.y = A[(size_t)row * lda + kk + 1];
  return a;
}

// B fragment, row-major KxN matrix, 4x16 tile at (k0,n0).
__device__ __forceinline__ v2f load_b_rm(const float* __restrict__ Bm, int ldb,
                                         int k0, int n0, int lane) {
  int col = n0 + (lane & 15);
  int kk  = k0 + ((lane >> 4) << 1);
  v2f b;
  b.x = Bm[(size_t)kk * ldb + col];
  b.y = Bm[(size_t)(kk + 1) * ldb + col];
  return b;
}

__device__ __forceinline__ float sigmoidf_(float v) { return 1.0f / (1.0f + expf(-v)); }
__device__ __forceinline__ float geluf_(float v) {
  return 0.5f * v * (1.0f + erff(v * 0.70710678118654752440f));
}

// 16-byte global -> LDS stage: async (ASYNCcnt) on CDNA5, sync fallback.
__device__ __forceinline__ void stage16(float* lds_dst, const float* gsrc) {
#if HAVE_ASYNC_LDS
  __builtin_amdgcn_global_load_async_to_lds_b128(
      (__attribute__((address_space(1))) v4i_*)(gsrc),
      (__attribute__((address_space(3))) v4i_*)(lds_dst), 0, 0);
#else
  *(float4*)lds_dst = *(const float4*)gsrc;
#endif
}

__device__ __forceinline__ void wait_async() {
#if HAVE_ASYNC_LDS
  __builtin_amdgcn_s_wait_asynccnt(0);
#endif
}

// ---------------------------------------------------------------------------
// L2 normalize rows of 512 (x -> x_n), 256 threads / row
// ---------------------------------------------------------------------------
__global__ void k_l2norm_rows512(const float* __restrict__ x, float* __restrict__ y) {
  __shared__ float red[256];
  int row = blockIdx.x, t = threadIdx.x;
  size_t base = (size_t)row * 512;
  float v0 = x[base + t], v1 = x[base + 256 + t];
  red[t] = v0 * v0 + v1 * v1;
  __syncthreads();
  for (int s = 128; s > 0; s >>= 1) {
    if (t < s) red[t] += red[t + s];
    __syncthreads();
  }
  float inv = 1.0f / fmaxf(sqrtf(red[0]), 1e-12f);
  y[base + t] = v0 * inv;
  y[base + 256 + t] = v1 * inv;
}

// L2 normalize patterns (64 x 512) and store transposed pT (512 x 64)
__global__ void k_norm_patternsT(const float* __restrict__ p, float* __restrict__ pT) {
  __shared__ float red[256];
  int n = blockIdx.x, t = threadIdx.x;
  size_t base = (size_t)n * 512;
  float v0 = p[base + t], v1 = p[base + 256 + t];
  red[t] = v0 * v0 + v1 * v1;
  __syncthreads();
  for (int s = 128; s > 0; s >>= 1) {
    if (t < s) red[t] += red[t + s];
    __syncthreads();
  }
  float inv = 1.0f / fmaxf(sqrtf(red[0]), 1e-12f);
  pT[(size_t)t * 64 + n] = v0 * inv;
  pT[(size_t)(t + 256) * 64 + n] = v1 * inv;
}

// L2 normalize templates (128 x 64) store transposed tT (64 x 128); 64 thr/row
__global__ void k_norm_templatesT(const float* __restrict__ tm, float* __restrict__ tT) {
  __shared__ float red[64];
  int p = blockIdx.x, t = threadIdx.x;
  float v = tm[(size_t)p * 64 + t];
  red[t] = v * v;
  __syncthreads();
  for (int s = 32; s > 0; s >>= 1) {
    if (t < s) red[t] += red[t + s];
    __syncthreads();
  }
  float inv = 1.0f / fmaxf(sqrtf(red[0]), 1e-12f);
  tT[(size_t)t * 128 + p] = v * inv;
}

// L2 normalize rows of 64 (input_acts -> a_n); 64 threads / row
__global__ void k_norm_rows64(const float* __restrict__ in, float* __restrict__ outp) {
  __shared__ float red[64];
  int row = blockIdx.x, t = threadIdx.x;
  float v = in[(size_t)row * 64 + t];
  red[t] = v * v;
  __syncthreads();
  for (int s = 32; s > 0; s >>= 1) {
    if (t < s) red[t] += red[t + s];
    __syncthreads();
  }
  float inv = 1.0f / fmaxf(sqrtf(red[0]), 1e-12f);
  outp[(size_t)row * 64 + t] = v * inv;
}

// ---------------------------------------------------------------------------
// input_acts = sigmoid(x_n @ p_nT)   (2048 x 64, K = 512)
// ---------------------------------------------------------------------------
__global__ void k_input_acts(const float* __restrict__ xn, const float* __restrict__ pT,
                             float* __restrict__ acts) {
  int lane = threadIdx.x & 31;
  int wave = threadIdx.x >> 5;
  int m0 = blockIdx.x * 128 + wave * 16;
  v8f acc[4] = {};
  for (int k = 0; k < 512; k += 4) {
    v2f a = load_a_rm(xn, 512, m0, k, lane);
#pragma unroll
    for (int j = 0; j < 4; ++j) {
      v2f b = load_b_rm(pT, 64, k, j * 16, lane);
      acc[j] = wmma4(a, b, acc[j]);
    }
  }
  int colb = lane & 15;
  int rbase = m0 + ((lane >> 4) << 3);
#pragma unroll
  for (int j = 0; j < 4; ++j) {
#pragma unroll
    for (int r = 0; r < 8; ++r)
      acts[(size_t)(rbase + r) * 64 + j * 16 + colb] = sigmoidf_(acc[j][r]);
  }
}

// ---------------------------------------------------------------------------
// hid = gelu(x @ fe_w1 + fe_b1)   (2048 x 2048, K = 512)
// ---------------------------------------------------------------------------
__global__ void k_hidden(const float* __restrict__ x, const float* __restrict__ w1,
                         const float* __restrict__ b1, float* __restrict__ hid) {
  int lane = threadIdx.x & 31;
  int wave = threadIdx.x >> 5;
  int m0 = blockIdx.y * 128 + wave * 16;
  int n0 = blockIdx.x * 64;
  v8f acc[4] = {};
  for (int k = 0; k < 512; k += 4) {
    v2f a = load_a_rm(x, 512, m0, k, lane);
#pragma unroll
    for (int j = 0; j < 4; ++j) {
      v2f b = load_b_rm(w1, 2048, k, n0 + j * 16, lane);
      acc[j] = wmma4(a, b, acc[j]);
    }
  }
  int colb = lane & 15;
  int rbase = m0 + ((lane >> 4) << 3);
#pragma unroll
  for (int j = 0; j < 4; ++j) {
    int col = n0 + j * 16 + colb;
    float bias = b1[col];
#pragma unroll
    for (int r = 0; r < 8; ++r)
      hid[(size_t)(rbase + r) * 2048 + col] = geluf_(acc[j][r] + bias);
  }
}

// ---------------------------------------------------------------------------
// Fused dominant kernel:
//   intermediate[t,h] = sum_n acts[t,n] * (hid[t,:] @ fe_w2[:, n*512+h] + b2[n,h])
// acts folded into A rows (diag(acts_n) * A) before the WMMA,
// bias term = acts @ b2.reshape(64,512) computed as WMMA prologue into acc.
// B tiles (fe_w2) double-buffered in LDS via GLOBAL_LOAD_ASYNC_TO_LDS_B128.
// WG = 256 thr / 8 waves -> 64 tokens x 64 h; wave = 16 tokens x 32 h.
// Grid: (512/64 h-tiles, 2048/64 m-tiles) = 256 WGs.
// ---------------------------------------------------------------------------
#define KT 32
#define LPITCH 68  // padded row pitch (floats): avoids LDS bank conflicts, 16B-aligned

__global__ void k_intermediate(const float* __restrict__ hid, const float* __restrict__ w2,
                               const float* __restrict__ b2, const float* __restrict__ acts,
                               float* __restrict__ inter) {
  __shared__ float lbs[2][KT * LPITCH];
  int lane = threadIdx.x & 31;
  int wave = threadIdx.x >> 5;
  int m0 = blockIdx.y * 64 + (wave >> 1) * 16;
  int h0 = blockIdx.x * 64;
  int hsub = (wave & 1) * 32;
  int arow = m0 + (lane & 15);

  // acc = bias term: acts(2048x64) @ b2 viewed as (64x512), K = 64
  v8f acc[2] = {};
  for (int k = 0; k < 64; k += 4) {
    v2f a = load_a_rm(acts, 64, m0, k, lane);
#pragma unroll
    for (int j = 0; j < 2; ++j) {
      v2f b = load_b_rm(b2, 512, k, h0 + hsub + j * 16, lane);
      acc[j] = wmma4(a, b, acc[j]);
    }
  }

  // staging role of this thread: two 16-byte pieces per KTx64 chunk
  int srow = threadIdx.x >> 4;        // 0..15
  int scol = (threadIdx.x & 15) * 4;  // 0..60

  for (int n = 0; n < 64; ++n) {
    float av = acts[(size_t)arow * 64 + n];
    const float* bsrc = w2 + (size_t)n * 512 + h0;  // row k stride = 32768

    // prefetch chunk 0
    {
      float* lb = lbs[0];
      stage16(lb + srow * LPITCH + scol, bsrc + (size_t)srow * 32768 + scol);
      stage16(lb + (srow + 16) * LPITCH + scol,
              bsrc + (size_t)(srow + 16) * 32768 + scol);
    }
    for (int kc = 0; kc < 2048 / KT; ++kc) {
      wait_async();
      __syncthreads();  // chunk kc resident; prev buffer free
      if (kc + 1 < 2048 / KT) {
        float* lb = lbs[(kc + 1) & 1];
        const float* g = bsrc + (size_t)(kc + 1) * KT * 32768;
        stage16(lb + srow * LPITCH + scol, g + (size_t)srow * 32768 + scol);
        stage16(lb + (srow + 16) * LPITCH + scol,
                g + (size_t)(srow + 16) * 32768 + scol);
      }
      const float* lb = lbs[kc & 1];
#pragma unroll
      for (int k = 0; k < KT; k += 4) {
        v2f a = load_a_rm(hid, 2048, m0, kc * KT + k, lane);
        a.x *= av;
        a.y *= av;
        int kk = k + ((lane >> 4) << 1);
#pragma unroll
        for (int j = 0; j < 2; ++j) {
          int col = hsub + j * 16 + (lane & 15);
          v2f b;
          b.x = lb[kk * LPITCH + col];
          b.y = lb[(kk + 1) * LPITCH + col];
          acc[j] = wmma4(a, b, acc[j]);
        }
      }
      __syncthreads();  // all waves done reading chunk kc
    }
  }

  int colb = lane & 15;
  int rbase = m0 + ((lane >> 4) << 3);
#pragma unroll
  for (int j = 0; j < 2; ++j) {
#pragma unroll
    for (int r = 0; r < 8; ++r)
      inter[(size_t)(rbase + r) * 512 + h0 + hsub + j * 16 + colb] = acc[j][r];
  }
}

// ---------------------------------------------------------------------------
// out = layer_norm(a + bsum) * g + bb      rows of 512
// ---------------------------------------------------------------------------
__global__ void k_add_ln(const float* __restrict__ a, const float* __restrict__ bsum,
                         const float* __restrict__ g, const float* __restrict__ bb,
                         float* __restrict__ out) {
  __shared__ float r1[256];
  __shared__ float r2[256];
  int row = blockIdx.x, t = threadIdx.x;
  size_t base = (size_t)row * 512;
  float v0 = a[base + t] + bsum[base + t];
  float v1 = a[base + 256 + t] + bsum[base + 256 + t];
  r1[t] = v0 + v1;
  r2[t] = v0 * v0 + v1 * v1;
  __syncthreads();
  for (int s = 128; s > 0; s >>= 1) {
    if (t < s) { r1[t] += r1[t + s]; r2[t] += r2[t + s]; }
    __syncthreads();
  }
  float mean = r1[0] * (1.0f / 512.0f);
  float var = r2[0] * (1.0f / 512.0f) - mean * mean;
  float rstd = rsqrtf(var + 1e-5f);
  out[base + t] = (v0 - mean) * rstd * g[t] + bb[t];
  out[base + 256 + t] = (v1 - mean) * rstd * g[t + 256] + bb[t + 256];
}

// ---------------------------------------------------------------------------
// proc_acts = sigmoid(a_n @ t_nT)   (2048 x 128, K = 64)
// ---------------------------------------------------------------------------
__global__ void k_proc_acts(const float* __restrict__ an, const float* __restrict__ tT,
                            float* __restrict__ proc) {
  int lane = threadIdx.x & 31;
  int wave = threadIdx.x >> 5;
  int m0 = blockIdx.y * 128 + wave * 16;
  int n0 = blockIdx.x * 64;
  v8f acc[4] = {};
  for (int k = 0; k < 64; k += 4) {
    v2f a = load_a_rm(an, 64, m0, k, lane);
#pragma unroll
    for (int j = 0; j < 4; ++j) {
      v2f b = load_b_rm(tT, 128, k, n0 + j * 16, lane);
      acc[j] = wmma4(a, b, acc[j]);
    }
  }
  int colb = lane & 15;
  int rbase = m0 + ((lane >> 4) << 3);
#pragma unroll
  for (int j = 0; j < 4; ++j) {
#pragma unroll
    for (int r = 0; r < 8; ++r)
      proc[(size_t)(rbase + r) * 128 + n0 + j * 16 + colb] = sigmoidf_(acc[j][r]);
  }
}

// norms[b,p] = sum_s proc[b,s,p] + 1e-8   (512 values; coalesced over p)
__global__ void k_procsum(const float* __restrict__ proc, float* __restrict__ norms) {
  int t = blockIdx.x * blockDim.x + threadIdx.x;
  if (t >= 512) return;
  int b = t >> 7, p = t & 127;
  float s = 0.0f;
  for (int srow = 0; srow < 512; ++srow)
    s += proc[((size_t)b * 512 + srow) * 128 + p];
  norms[t] = s + 1e-8f;
}

// ---------------------------------------------------------------------------
// contexts[b,p,h] = (sum_s proc[b,s,p] * x1[b,s,h]) / norms[b,p]
// ---------------------------------------------------------------------------
__global__ void k_contexts(const float* __restrict__ proc, const float* __restrict__ x1,
                           const float* __restrict__ norms, float* __restrict__ ctx) {
  int lane = threadIdx.x & 31;
  int wave = threadIdx.x >> 5;
  int b = blockIdx.z;
  const float* Pb = proc + (size_t)b * 512 * 128;
  const float* Xb = x1 + (size_t)b * 512 * 512;
  int m0 = wave * 16;  // 8 waves cover all 128 p
  int n0 = blockIdx.x * 64;
  v8f acc[4] = {};
  for (int k = 0; k < 512; k += 4) {
    // A = proc^T: element (p=row, s=k) lives at Pb[s*128 + p]
    int row = m0 + (lane & 15);
    int kk = k + ((lane >> 4) << 1);
    v2f a;
    a.x = Pb[(size_t)kk * 128 + row];
    a.y = Pb[(size_t)(kk + 1) * 128 + row];
#pragma unroll
    for (int j = 0; j < 4; ++j) {
      v2f bf = load_b_rm(Xb, 512, k, n0 + j * 16, lane);
      acc[j] = wmma4(a, bf, acc[j]);
    }
  }
  int colb = lane & 15;
  int rbase = m0 + ((lane >> 4) << 3);
#pragma unroll
  for (int r = 0; r < 8; ++r) {
    float inv = 1.0f / norms[b * 128 + rbase + r];
#pragma unroll
    for (int j = 0; j < 4; ++j)
      ctx[((size_t)b * 128 + rbase + r) * 512 + n0 + j * 16 + colb] = acc[j][r] * inv;
  }
}

// ---------------------------------------------------------------------------
// transformed[b,p,k] = sum_h ctx[b,p,h] * wt[p,k,h]  (memory-bound on wt)
// ---------------------------------------------------------------------------
__global__ void k_transform(const float* __restrict__ ctx, const float* __restrict__ wt,
                            float* __restrict__ tr) {
  int wid = blockIdx.x * 8 + (threadIdx.x >> 5);
  int lane = threadIdx.x & 31;
  int p = wid >> 9;
  int kk = wid & 511;
  const float4* w4 = (const float4*)(wt + ((size_t)p * 512 + kk) * 512);
  float s0 = 0.f, s1 = 0.f, s2 = 0.f, s3 = 0.f;
#pragma unroll
  for (int i = 0; i < 4; ++i) {
    float4 w = w4[lane * 4 + i];
    const float4* c0 = (const float4*)(ctx + ((size_t)0 * 128 + p) * 512);
    const float4* c1 = (const float4*)(ctx + ((size_t)1 * 128 + p) * 512);
    const float4* c2 = (const float4*)(ctx + ((size_t)2 * 128 + p) * 512);
    const float4* c3 = (const float4*)(ctx + ((size_t)3 * 128 + p) * 512);
    float4 a0 = c0[lane * 4 + i], a1 = c1[lane * 4 + i];
    float4 a2 = c2[lane * 4 + i], a3 = c3[lane * 4 + i];
    s0 += w.x * a0.x + w.y * a0.y + w.z * a0.z + w.w * a0.w;
    s1 += w.x * a1.x + w.y * a1.y + w.z * a1.z + w.w * a1.w;
    s2 += w.x * a2.x + w.y * a2.y + w.z * a2.z + w.w * a2.w;
    s3 += w.x * a3.x + w.y * a3.y + w.z * a3.z + w.w * a3.w;
  }
  for (int off = 16; off > 0; off >>= 1) {
    s0 += __shfl_xor(s0, off, 32);
    s1 += __shfl_xor(s1, off, 32);
    s2 += __shfl_xor(s2, off, 32);
    s3 += __shfl_xor(s3, off, 32);
  }
  if (lane == 0) {
    tr[((size_t)0 * 128 + p) * 512 + kk] = s0;
    tr[((size_t)1 * 128 + p) * 512 + kk] = s1;
    tr[((size_t)2 * 128 + p) * 512 + kk] = s2;
    tr[((size_t)3 * 128 + p) * 512 + kk] = s3;
  }
}

// ---------------------------------------------------------------------------
// combined[b,s,h] = sum_p transformed[b,p,h] * proc[b,s,p]   (K = 128)
// ---------------------------------------------------------------------------
__global__ void k_combined(const float* __restrict__ proc, const float* __restrict__ tr,
                           float* __restrict__ comb) {
  int lane = threadIdx.x & 31;
  int wave = threadIdx.x >> 5;
  int b = blockIdx.z;
  const float* A = proc + (size_t)b * 512 * 128;  // S x P row-major
  const float* Bm = tr + (size_t)b * 128 * 512;   // P x H row-major
  int m0 = blockIdx.y * 128 + wave * 16;
  int n0 = blockIdx.x * 64;
  v8f acc[4] = {};
  for (int k = 0; k < 128; k += 4) {
    v2f a = load_a_rm(A, 128, m0, k, lane);
#pragma unroll
    for (int j = 0; j < 4; ++j) {
      v2f bf = load_b_rm(Bm, 512, k, n0 + j * 16, lane);
      acc[j] = wmma4(a, bf, acc[j]);
    }
  }
  int colb = lane & 15;
  int rbase = m0 + ((lane >> 4) << 3);
#pragma unroll
  for (int j = 0; j < 4; ++j) {
#pragma unroll
    for (int r = 0; r < 8; ++r)
      comb[((size_t)b * 512 + rbase + r) * 512 + n0 + j * 16 + colb] = acc[j][r];
  }
}

// ---------------------------------------------------------------------------
extern "C" void kernel_launch(void* const* d_in, const int* in_sizes, int n_in,
                              void* d_out, int out_size, void* d_ws, size_t ws_size,
                              hipStream_t stream) {
  const float* x     = (const float*)d_in[0];   // (4,512,512)
  const float* pat   = (const float*)d_in[1];   // (64,512)
  const float* fe_w1 = (const float*)d_in[2];   // (512,2048)
  const float* fe_b1 = (const float*)d_in[3];   // (2048)
  const float* fe_w2 = (const float*)d_in[4];   // (2048,32768)
  const float* fe_b2 = (const float*)d_in[5];   // (32768)
  const float* tmpl  = (const float*)d_in[6];   // (128,64)
  const float* wt    = (const float*)d_in[7];   // (128,512,512)
  const float* ln1_g = (const float*)d_in[8];
  const float* ln1_b = (const float*)d_in[9];
  const float* ln2_g = (const float*)d_in[10];
  const float* ln2_b = (const float*)d_in[11];

  float* out  = (float*)d_out;                  // (4,512,512) = 1048576
  float* acts = out + 1048576;                  // (4,512,64)  = 131072
  float* proc = out + 1179648;                  // (4,512,128) = 262144

  float* ws = (float*)d_ws;
  float* xn    = ws;               // 1048576
  float* pT    = xn + 1048576;     // 32768  (512 x 64)
  float* tT    = pT + 32768;       // 8192   (64 x 128)
  float* hid   = tT + 8192;        // 4194304 (2048 x 2048)
  float* inter = hid + 4194304;    // 1048576
  float* x1    = inter + 1048576;  // 1048576
  float* an    = x1 + 1048576;     // 131072
  float* norms = an + 131072;      // 512
  float* ctx   = norms + 512;      // 262144 (4 x 128 x 512)
  float* tr    = ctx + 262144;     // 262144
  float* comb  = tr + 262144;      // 1048576

  // ---- InputNeurons ----
  k_l2norm_rows512<<<2048, 256, 0, stream>>>(x, xn);
  k_norm_patternsT<<<64, 256, 0, stream>>>(pat, pT);
  k_input_acts<<<16, 256, 0, stream>>>(xn, pT, acts);
  k_hidden<<<dim3(32, 16), 256, 0, stream>>>(x, fe_w1, fe_b1, hid);
  k_intermediate<<<dim3(8, 32), 256, 0, stream>>>(hid, fe_w2, fe_b2, acts, inter);
  k_add_ln<<<2048, 256, 0, stream>>>(x, inter, ln1_g, ln1_b, x1);

  // ---- ProcessNeurons ----
  k_norm_templatesT<<<128, 64, 0, stream>>>(tmpl, tT);
  k_norm_rows64<<<2048, 64, 0, stream>>>(acts, an);
  k_proc_acts<<<dim3(2, 16), 256, 0, stream>>>(an, tT, proc);
  k_procsum<<<2, 256, 0, stream>>>(proc, norms);
  k_contexts<<<dim3(8, 1, 4), 256, 0, stream>>>(proc, x1, norms, ctx);
  k_transform<<<8192, 256, 0, stream>>>(ctx, wt, tr);
  k_combined<<<dim3(8, 4, 4), 256, 0, stream>>>(proc, tr, comb);
  k_add_ln<<<2048, 256, 0, stream>>>(x1, comb, ln2_g, ln2_b, out);
}